// PostProcessViz_20933670600918
// MI455X (gfx1250) — compile-verified
//
#include <hip/hip_runtime.h>
#include <math.h>
#include <stdint.h>

typedef float v2f __attribute__((ext_vector_type(2)));
typedef float v4f __attribute__((ext_vector_type(4)));

#define NB 32
#define NQ 1000
#define NC 1200
#define QPAD 1024

// ---------------------------------------------------------------------------
// Kernel 1: human-score filter + box convert/scale.  [B*Q] threads.
// ---------------------------------------------------------------------------
__global__ __launch_bounds__(256)
void pp_prep(const float* __restrict__ hl, const float* __restrict__ pb,
             const int* __restrict__ ih, const int* __restrict__ iw,
             float* __restrict__ ws_boxes, float* __restrict__ ws_score,
             float* __restrict__ out_boxes)
{
    int i = blockIdx.x * blockDim.x + threadIdx.x;
    if (i >= NB * NQ) return;

    v2f l = ((const v2f*)hl)[i];
    float d = l[0] - l[1];
    // 2-class softmax max prob = sigmoid(|l0-l1|); argmax==0 iff l0>=l1 (stable tie)
    float score = 1.0f / (1.0f + expf(-fabsf(d)));
    bool keep0 = (d >= 0.0f) && (score >= 0.7f);

    float W = (float)iw[0], H = (float)ih[0];
    v4f b = ((const v4f*)pb)[i];
    v4f box;
    box[0] = (b[0] - 0.5f * b[2]) * W;
    box[1] = (b[1] - 0.5f * b[3]) * H;
    box[2] = (b[0] + 0.5f * b[2]) * W;
    box[3] = (b[1] + 0.5f * b[3]) * H;
    ((v4f*)ws_boxes)[i] = box;

    v4f bi;
    bi[0] = (float)(int)box[0];  // astype(int32): truncate toward zero
    bi[1] = (float)(int)box[1];
    bi[2] = (float)(int)box[2];
    bi[3] = (float)(int)box[3];
    ((v4f*)out_boxes)[i] = bi;

    ws_score[i] = keep0 ? score : -INFINITY;   // -inf == invalid for NMS
}

// ---------------------------------------------------------------------------
// Kernel 2: per-image greedy NMS.  One 1024-thread block (32 wave32) per image.
// ---------------------------------------------------------------------------
__global__ __launch_bounds__(1024)
void pp_nms(const float* __restrict__ ws_boxes, const float* __restrict__ ws_score,
            float* __restrict__ out_keep)
{
    __shared__ float        skey[QPAD];
    __shared__ int          sidx[QPAD];
    __shared__ v4f          sbox[QPAD];
    __shared__ unsigned int keepw[QPAD / 32];
    __shared__ unsigned int cmask[32][33];      // +1 pad: conflict-free stores

    const int b = blockIdx.x;
    const int t = threadIdx.x;

    skey[t] = (t < NQ) ? ws_score[b * NQ + t] : -INFINITY;
    sidx[t] = t;

    // ---- bitonic sort, descending by masked score, index payload ----
    for (int k = 2; k <= QPAD; k <<= 1)
        for (int j = k >> 1; j > 0; j >>= 1) {
            __syncthreads();
            int ixj = t ^ j;
            if (ixj > t) {
                float a = skey[t], c2 = skey[ixj];
                bool desc = ((t & k) == 0);
                if (desc ? (a < c2) : (a > c2)) {
                    skey[t] = c2; skey[ixj] = a;
                    int tmp = sidx[t]; sidx[t] = sidx[ixj]; sidx[ixj] = tmp;
                }
            }
        }
    __syncthreads();

    // ---- CDNA5 async gather: boxes (sorted order) -> LDS, ASYNCcnt tracked ----
    {
        int src = sidx[t];
        if (src >= NQ) src = 0;                       // pad slots: harmless dup
        const v4f* gp = (const v4f*)ws_boxes + ((size_t)b * NQ + src);
        unsigned long long ga = (unsigned long long)(uintptr_t)gp;
        unsigned int lo = (unsigned int)(uintptr_t)&sbox[t];  // LDS byte offset
        asm volatile("global_load_async_to_lds_b128 %0, %1, off"
                     :: "v"(lo), "v"(ga) : "memory");
        asm volatile("s_wait_asynccnt 0" ::: "memory");
    }
    __syncthreads();

    // ---- init keep bitset: bit = "box valid" (finite masked score) ----
    if (t < QPAD / 32) {
        unsigned int wrd = 0u;
        for (int bit = 0; bit < 32; ++bit)
            if (skey[t * 32 + bit] > -INFINITY) wrd |= (1u << bit);
        keepw[t] = wrd;
    }
    __syncthreads();

    // ---- chunked suppression: 32 rows/chunk; lane=row, word=tid>>5 ----
    const int r = t & 31;      // row within chunk (lane-major -> sbox[j] broadcast)
    const int w = t >> 5;      // 32-column word owned by this thread
    for (int c = 0; c < QPAD; c += 32) {
        unsigned int rowword = keepw[c >> 5];          // uniform across block
        if (rowword != 0u) {                           // skip fully-dead chunks
            int i = c + r;
            unsigned int m = 0u;
            if (i < NQ && ((rowword >> r) & 1u)) {     // only live rows suppress
                v4f bi = sbox[i];
                float areai = (bi[2] - bi[0]) * (bi[3] - bi[1]);
                int jbase = w * 32;
                #pragma unroll 4
                for (int jj = 0; jj < 32; ++jj) {
                    int j = jbase + jj;
                    if (j > i) {
                        v4f bj = sbox[j];
                        float xx1 = fmaxf(bi[0], bj[0]);
                        float yy1 = fmaxf(bi[1], bj[1]);
                        float xx2 = fminf(bi[2], bj[2]);
                        float yy2 = fminf(bi[3], bj[3]);
                        float iw_ = fmaxf(xx2 - xx1, 0.0f);
                        float ih_ = fmaxf(yy2 - yy1, 0.0f);
                        float inter = iw_ * ih_;
                        float areaj = (bj[2] - bj[0]) * (bj[3] - bj[1]);
                        float iou = inter / (areai + areaj - inter + 1e-9f);
                        if (iou > 0.5f) m |= (1u << jj);
                    }
                }
            }
            cmask[r][w] = m;
            __syncthreads();
            // greedy scan of this chunk by one wave32: lane owns one keep word
            if (t < 32) {
                int cw = c >> 5;
                for (int rr = 0; rr < 32; ++rr) {
                    unsigned int alive = keepw[cw];        // in-order LDS per wave
                    if ((alive >> rr) & 1u)
                        keepw[t] &= ~cmask[rr][t];         // j>i bits only
                }
            }
            __syncthreads();
        }
    }

    __syncthreads();
    // ---- scatter keep back to original box order (also final output #4) ----
    {
        int orig = sidx[t];
        if (orig < NQ) {
            unsigned int kb = (keepw[t >> 5] >> (t & 31)) & 1u;
            out_keep[b * NQ + orig] = kb ? 1.0f : 0.0f;
        }
    }
}

// ---------------------------------------------------------------------------
// Kernel 3: streaming multi-label extraction over [B,Q,C].  ~460 MB traffic,
// HBM-bound (~20 us at 23.3 TB/s).  Non-temporal B128 loads/stores.
// ---------------------------------------------------------------------------
__global__ __launch_bounds__(256)
void pp_extract(const float* __restrict__ logits, const float* __restrict__ keepf,
                float* __restrict__ out_s, float* __restrict__ out_m)
{
    const int N4 = NB * NQ * NC / 4;       // 9,600,000 float4s
    const unsigned int VPR = NC / 4;       // 300 vec4 per (b,q) row
    int stride = gridDim.x * blockDim.x;
    for (int i4 = blockIdx.x * blockDim.x + threadIdx.x; i4 < N4; i4 += stride) {
        unsigned int row = (unsigned int)i4 / VPR;     // magic-number div
        const v4f* lp = (const v4f*)logits + i4;
        int pf = i4 + stride; if (pf > N4 - 1) pf = N4 - 1;
        __builtin_prefetch((const v4f*)logits + pf, 0, 0);  // global_prefetch_b8
        v4f x = __builtin_nontemporal_load(lp);
        float k = keepf[row];                          // broadcast within row
        bool kb = (k != 0.0f);
        v4f s, m;
        #pragma unroll
        for (int c = 0; c < 4; ++c) {
            bool on = kb && (x[c] >= 0.25f);
            s[c] = on ? (x[c] + 1.0f) * 0.5f : 0.0f;
            m[c] = on ? 1.0f : 0.0f;
        }
        __builtin_nontemporal_store(s, (v4f*)out_s + i4);
        __builtin_nontemporal_store(m, (v4f*)out_m + i4);
    }
}

// ---------------------------------------------------------------------------
extern "C" void kernel_launch(void* const* d_in, const int* in_sizes, int n_in,
                              void* d_out, int out_size, void* d_ws, size_t ws_size,
                              hipStream_t stream)
{
    const float* hl = (const float*)d_in[0];   // human_logits [B,Q,2]
    const float* pl = (const float*)d_in[1];   // pred_logits  [B,Q,C]
    const float* pb = (const float*)d_in[2];   // pred_boxes   [B,Q,4]
    const int*   ih = (const int*)d_in[3];     // imgsize_h
    const int*   iw = (const int*)d_in[4];     // imgsize_w

    float* out        = (float*)d_out;
    float* out_scores = out;                                  // B*Q*C
    float* out_mask   = out + (size_t)NB * NQ * NC;           // B*Q*C
    float* out_boxes  = out + 2 * (size_t)NB * NQ * NC;       // B*Q*4
    float* out_keep   = out_boxes + (size_t)NB * NQ * 4;      // B*Q

    float* ws       = (float*)d_ws;
    float* ws_boxes = ws;                    // B*Q*4 floats (512 KB)
    float* ws_score = ws + NB * NQ * 4;      // B*Q   floats (128 KB)

    pp_prep<<<(NB * NQ + 255) / 256, 256, 0, stream>>>(
        hl, pb, ih, iw, ws_boxes, ws_score, out_boxes);

    pp_nms<<<NB, 1024, 0, stream>>>(ws_boxes, ws_score, out_keep);

    pp_extract<<<9600, 256, 0, stream>>>(pl, out_keep, out_scores, out_mask);
}